// Swin_axial_85169201480313
// MI455X (gfx1250) — compile-verified
//
#include <hip/hip_runtime.h>
#include <math.h>

typedef __attribute__((ext_vector_type(2))) float v2f;
typedef __attribute__((ext_vector_type(8))) float v8f;

#define WMMA_F32(a, b, c) \
  __builtin_amdgcn_wmma_f32_16x16x4_f32(false, (a), false, (b), (short)0, (c), false, false)

constexpr int Cdim   = 256;   // channels
constexpr int H      = 44;    // sequence length / patch side
constexpr int HP     = 48;    // padded to 3 WMMA tiles
constexpr int CWP    = 52;    // conv LDS row stride (halo-safe)
constexpr int NPATCH = 64;    // 16 * 4 patches
constexpr int BBATCH = 2816;  // NPATCH * 44
constexpr int GP     = 64;    // C / GROUPS

// ---------------------------------------------------------------------------
// Axial attention pass. One workgroup (8 waves) per batch row bb in [0,2816).
// pass==0: height attention, reads raw (16,256,88,88) input with patch math,
//          writes (NP,C,44,44).
// pass==1: width attention, reads/writes (NP,C,44,44).
// LDS: sx[256][48] sqkv[512][48] srel[128][88] ssim[48][48] + bn tables
// ---------------------------------------------------------------------------
__global__ __launch_bounds__(256) void axial_kernel(
    const float* __restrict__ xin, const float* __restrict__ qkvw,
    const float* __restrict__ bnqkv, const float* __restrict__ bnsim,
    const float* __restrict__ bnout, const float* __restrict__ rel,
    float* __restrict__ out, int pass) {
  extern __shared__ float smem[];
  float* sx    = smem;                  // 256*48
  float* sqkv  = sx + Cdim * HP;        // 512*48
  float* srel  = sqkv + 2 * Cdim * HP;  // 128*88
  float* ssim  = srel + 128 * 88;       // 48*48
  float* sbnqs = ssim + HP * HP;        // 512 qkv-bn scale
  float* sbnqb = sbnqs + 512;           // 512 qkv-bn shift
  float* sbnos = sbnqb + 512;           // 512 out-bn scale
  float* sbnob = sbnos + 512;           // 512 out-bn shift

  const int tid  = threadIdx.x;
  const int lane = tid & 31;
  const int wave = tid >> 5;
  const int lr   = lane & 15;  // A: M row / B,C,D: N col
  const int hi   = lane >> 4;  // half-wave select

  const int bb    = blockIdx.x;
  const int patch = bb / H;
  const int col   = bb - patch * H;
  const int n = patch >> 2, pa = (patch >> 1) & 1, pb = patch & 1;

  // ---- stage activation slice x(bb, c=0..255, i=0..43) into LDS -----------
  for (int idx = tid; idx < Cdim * H; idx += 256) {
    int c = idx / H, i = idx - c * H;
    long addr;
    if (pass == 0)
      addr = (((long)(n * Cdim + c) * 88) + (pa * H + i)) * 88 + (pb * H + col);
    else
      addr = (((long)(patch * Cdim + c) * H) + col) * H + i;
    sx[c * HP + i] = xin[addr];
  }
  for (int idx = tid; idx < Cdim * (HP - H); idx += 256) {
    int c = idx >> 2, i = H + (idx & 3);
    sx[c * HP + i] = 0.f;  // zero-pad K columns
  }
  // ---- stage rel table (128 x 87) ----------------------------------------
  for (int idx = tid; idx < 128 * 87; idx += 256) {
    int c = idx / 87, d = idx - c * 87;
    srel[c * 88 + d] = rel[idx];
  }
  // ---- precompute BN affine tables (512 channels each) --------------------
  for (int o = tid; o < 512; o += 256) {
    float sc = bnqkv[o] * rsqrtf(bnqkv[1536 + o] + 1e-5f);
    sbnqs[o] = sc;
    sbnqb[o] = bnqkv[512 + o] - bnqkv[1024 + o] * sc;
    float so = bnout[o] * rsqrtf(bnout[1536 + o] + 1e-5f);
    sbnos[o] = so;
    sbnob[o] = bnout[512 + o] - bnout[1024 + o] * so;
  }
  __syncthreads();

  // ---- QKV: (512x256) * (256x48) -> sqkv, bn_qkv fused --------------------
  // Register-blocked 1x3: one divergent A (weight) load feeds 3 WMMAs.
  for (int mt = wave; mt < 32; mt += 8) {
    v8f acc0 = {}, acc1 = {}, acc2 = {};
    const int o_a = (mt * 16 + lr) * Cdim;  // weight row for A lanes
    for (int kb = 0; kb < Cdim; kb += 4) {
      int k = kb + hi * 2;
      v2f a, b0, b1, b2;
      a.x = qkvw[o_a + k];
      a.y = qkvw[o_a + k + 1];
      b0.x = sx[k * HP + lr];
      b0.y = sx[(k + 1) * HP + lr];
      b1.x = sx[k * HP + 16 + lr];
      b1.y = sx[(k + 1) * HP + 16 + lr];
      b2.x = sx[k * HP + 32 + lr];
      b2.y = sx[(k + 1) * HP + 32 + lr];
      acc0 = WMMA_F32(a, b0, acc0);
      acc1 = WMMA_F32(a, b1, acc1);
      acc2 = WMMA_F32(a, b2, acc2);
    }
#pragma unroll
    for (int r = 0; r < 8; r++) {
      int o = mt * 16 + r + 8 * hi;
      float sc = sbnqs[o], sh = sbnqb[o];
      sqkv[o * HP + lr]      = acc0[r] * sc + sh;
      sqkv[o * HP + 16 + lr] = acc1[r] * sc + sh;
      sqkv[o * HP + 32 + lr] = acc2[r] * sc + sh;
    }
  }
  __syncthreads();

  // ---- per-group attention ------------------------------------------------
  for (int g = 0; g < 4; g++) {
    float aa[3], ab[3];  // bn_sim affine for qk / qr / kr channels
#pragma unroll
    for (int u = 0; u < 3; u++) {
      int ch = u * 4 + g;
      float sc = bnsim[ch] * rsqrtf(bnsim[36 + ch] + 1e-5f);
      aa[u] = sc;
      ab[u] = bnsim[12 + ch] - bnsim[24 + ch] * sc;
    }
    const float* q = sqkv + (g * 128) * HP;
    const float* k = sqkv + (g * 128 + 32) * HP;
    const float* v = sqkv + (g * 128 + 64) * HP;

    // qk = q^T(48x32) * k(32x48); qr/kr via VALU; bn_sim fused -> ssim
    for (int t = wave; t < 9; t += 8) {
      int mt = t / 3, nt = t - mt * 3;
      v8f acc = {};
      const int m_a = mt * 16 + lr;  // i
      const int n_b = nt * 16 + lr;  // j
      for (int kb = 0; kb < 32; kb += 4) {
        v2f a, b;
        int kc = kb + hi * 2;
        a.x = q[kc * HP + m_a];
        a.y = q[(kc + 1) * HP + m_a];
        b.x = k[kc * HP + n_b];
        b.y = k[(kc + 1) * HP + n_b];
        acc = WMMA_F32(a, b, acc);
      }
      int j = n_b;
#pragma unroll
      for (int r = 0; r < 8; r++) {
        int i = mt * 16 + r + 8 * hi;
        float val = 0.f;
        if (i < H && j < H) {
          float qr = 0.f, kr = 0.f;
          const float* rq = srel + (i - j + 43);
          const float* rk = srel + 32 * 88 + (j - i + 43);
          for (int c = 0; c < 32; c++) {
            qr += q[c * HP + i] * rq[c * 88];
            kr += k[c * HP + j] * rk[c * 88];
          }
          val = acc[r] * aa[0] + 0.1f * qr * aa[1] + 0.1f * kr * aa[2] +
                ab[0] + ab[1] + ab[2];
        }
        ssim[i * HP + j] = val;
      }
    }
    __syncthreads();

    // row softmax over j, zero the padding
    if (tid < HP) {
      int i = tid;
      if (i < H) {
        float mx = -1e30f;
        for (int j = 0; j < H; j++) mx = fmaxf(mx, ssim[i * HP + j]);
        float s = 0.f;
        for (int j = 0; j < H; j++) {
          float e = __expf(ssim[i * HP + j] - mx);
          ssim[i * HP + j] = e;
          s += e;
        }
        float inv = 1.f / s;
        for (int j = 0; j < H; j++) ssim[i * HP + j] *= inv;
        for (int j = H; j < HP; j++) ssim[i * HP + j] = 0.f;
      } else {
        for (int j = 0; j < HP; j++) ssim[i * HP + j] = 0.f;
      }
    }
    __syncthreads();

    // sv = v(64x48) * sim^T(48x48); sve via VALU; bn_out + pair-sum fused
    for (int t = wave; t < 12; t += 8) {
      int mt = t / 3, nt = t - mt * 3;
      v8f acc = {};
      const int m_a = mt * 16 + lr;  // c
      const int n_b = nt * 16 + lr;  // i
      for (int kb = 0; kb < HP; kb += 4) {
        v2f a, b;
        int kj = kb + hi * 2;
        a.x = v[m_a * HP + kj];
        a.y = v[m_a * HP + kj + 1];
        b.x = ssim[n_b * HP + kj];
        b.y = ssim[n_b * HP + kj + 1];
        acc = WMMA_F32(a, b, acc);
      }
      int i = n_b;
      if (i < H) {
        const float* srow = ssim + i * HP;
#pragma unroll
        for (int r = 0; r < 8; r++) {
          int c = mt * 16 + r + 8 * hi;
          float sve = 0.f;
          const float* rv = srel + (64 + c) * 88 + (i + 43);
          for (int j = 0; j < H; j++) sve += srow[j] * rv[-j];
          sve *= 0.1f;
          int cc = g * GP + c;
          int oe = 2 * cc, oo = oe + 1;
          float val = acc[r] * sbnos[oe] + sbnob[oe] + sve * sbnos[oo] + sbnob[oo];
          long addr;
          if (pass == 0)
            addr = (((long)(patch * Cdim + cc) * H) + i) * H + col;
          else
            addr = (((long)(patch * Cdim + cc) * H) + col) * H + i;
          out[addr] = val;
        }
      }
    }
    __syncthreads();  // ssim reused next group
  }
}

// ---------------------------------------------------------------------------
// 3x3 conv, residual fused on load, bias + unpatchify fused on store.
// One workgroup per (patch, output row y). Implicit GEMM with fp32 WMMA:
// M = 16 output channels, N = 16 x-pixels, K = 4 input channels per tap.
// Register-blocked 1x3 so each divergent weight load feeds 3 WMMAs.
// LDS: sin[3][256][52] (x padded: px = x+1, zero halo through px=51)
// ---------------------------------------------------------------------------
__global__ __launch_bounds__(256) void conv_kernel(
    const float* __restrict__ pin,   // (NP,C,44,44) attention output
    const float* __restrict__ xin,   // raw input for residual
    const float* __restrict__ wconv, // (256,256,3,3)
    const float* __restrict__ bconv, // (256,)
    float* __restrict__ outp) {      // (16,256,88,88)
  extern __shared__ float smem[];
  const int tid = threadIdx.x, lane = tid & 31, wave = tid >> 5;
  const int lr = lane & 15, hi = lane >> 4;
  const int blk = blockIdx.x;
  const int np = blk / H, y = blk - np * H;
  const int n = np >> 2, pa = (np >> 1) & 1, pb = np & 1;

  for (int row = 0; row < 3; row++) {
    float* s = smem + row * Cdim * CWP;
    int yy = y + row - 1;
    if (yy < 0 || yy >= H) {
      for (int idx = tid; idx < Cdim * CWP; idx += 256) s[idx] = 0.f;
    } else {
      for (int idx = tid; idx < Cdim * H; idx += 256) {
        int ic = idx / H, x = idx - ic * H;
        long ap = (((long)(np * Cdim + ic) * H) + yy) * H + x;
        long ar = (((long)(n * Cdim + ic) * 88) + pa * H + yy) * 88 + pb * H + x;
        s[ic * CWP + x + 1] = pin[ap] + xin[ar];
      }
      for (int idx = tid; idx < Cdim * 8; idx += 256) {
        int ic = idx >> 3, e = idx & 7;
        int px = (e == 0) ? 0 : (44 + e);  // px = 0, 45..51
        s[ic * CWP + px] = 0.f;
      }
    }
  }
  __syncthreads();

  for (int mt = wave; mt < 16; mt += 8) {
    v8f acc0 = {}, acc1 = {}, acc2 = {};
    for (int ky = 0; ky < 3; ky++) {
      const float* s = smem + ky * Cdim * CWP;
      for (int kx = 0; kx < 3; kx++) {
        const float* wl = wconv + (long)(mt * 16 + lr) * (Cdim * 9) + ky * 3 + kx;
        const float* sb = s + lr + kx;
        for (int kb = 0; kb < Cdim; kb += 4) {
          int ic = kb + hi * 2;
          v2f a, b0, b1, b2;
          a.x = wl[(long)ic * 9];
          a.y = wl[(long)(ic + 1) * 9];
          b0.x = sb[ic * CWP];
          b0.y = sb[(ic + 1) * CWP];
          b1.x = sb[ic * CWP + 16];
          b1.y = sb[(ic + 1) * CWP + 16];
          b2.x = sb[ic * CWP + 32];
          b2.y = sb[(ic + 1) * CWP + 32];
          acc0 = WMMA_F32(a, b0, acc0);
          acc1 = WMMA_F32(a, b1, acc1);
          acc2 = WMMA_F32(a, b2, acc2);
        }
      }
    }
#pragma unroll
    for (int r = 0; r < 8; r++) {
      int oc = mt * 16 + r + 8 * hi;
      float bias = bconv[oc];
      long ab_ = (((long)(n * Cdim + oc) * 88) + pa * H + y) * 88 + pb * H;
      outp[ab_ + lr] = acc0[r] + bias;            // x = lr        (< 16)
      outp[ab_ + 16 + lr] = acc1[r] + bias;       // x = 16 + lr   (< 32)
      if (32 + lr < H) outp[ab_ + 32 + lr] = acc2[r] + bias;  // x = 32+lr < 44
    }
  }
}

extern "C" void kernel_launch(void* const* d_in, const int* in_sizes, int n_in,
                              void* d_out, int out_size, void* d_ws,
                              size_t ws_size, hipStream_t stream) {
  (void)in_sizes; (void)n_in; (void)out_size;
  const float* x     = (const float*)d_in[0];
  const float* hqkvw = (const float*)d_in[1];
  const float* hbnq  = (const float*)d_in[2];
  const float* hbns  = (const float*)d_in[3];
  const float* hbno  = (const float*)d_in[4];
  const float* hrel  = (const float*)d_in[5];
  const float* wqkvw = (const float*)d_in[6];
  const float* wbnq  = (const float*)d_in[7];
  const float* wbns  = (const float*)d_in[8];
  const float* wbno  = (const float*)d_in[9];
  const float* wrel  = (const float*)d_in[10];
  const float* cw    = (const float*)d_in[11];
  const float* cb    = (const float*)d_in[12];
  float* out = (float*)d_out;

  const size_t bufelems = (size_t)NPATCH * Cdim * H * H;  // 31.7M floats
  float* buf_a = (float*)d_ws;
  float* buf_b = buf_a + bufelems;
  if (ws_size < 2 * bufelems * sizeof(float)) return;  // needs ~254 MB

  const size_t lds_axial =
      (size_t)(Cdim * HP + 2 * Cdim * HP + 128 * 88 + HP * HP + 4 * 512) *
      sizeof(float);
  const size_t lds_conv = (size_t)(3 * Cdim * CWP) * sizeof(float);

  axial_kernel<<<dim3(BBATCH), dim3(256), lds_axial, stream>>>(
      x, hqkvw, hbnq, hbns, hbno, hrel, buf_a, 0);
  axial_kernel<<<dim3(BBATCH), dim3(256), lds_axial, stream>>>(
      buf_a, wqkvw, wbnq, wbns, wbno, wrel, buf_b, 1);
  conv_kernel<<<dim3(BBATCH), dim3(256), lds_conv, stream>>>(
      buf_b, x, cw, cb, out);
}